// GATRegressor_43147241455944
// MI455X (gfx1250) — compile-verified
//
#include <hip/hip_runtime.h>
#include <hip/hip_bf16.h>

#define NN    100000
#define EE    400000
#define ETOT  (EE + NN)
#define INC   64
#define HIDC  64
#define EDIM  12
#define NHEAD 4
#define NLAY  2
#define NGR   4096
#define CEILDIV(a,b) (((a)+(b)-1)/(b))

typedef float v2f __attribute__((ext_vector_type(2)));
typedef float v8f __attribute__((ext_vector_type(8)));

__device__ __forceinline__ float elu1(float v) { return v > 0.f ? v : (expf(v) - 1.f); }
__device__ __forceinline__ int   encf(float f) { int b = __float_as_int(f); return b ^ ((b >> 31) & 0x7FFFFFFF); }
__device__ __forceinline__ float decf(int e)   { return __int_as_float(e ^ ((e >> 31) & 0x7FFFFFFF)); }

// ---------------- WMMA fp32 GEMM: C[M, NT*16] = act(A[M,64] @ B[64, NT*16] + bias) ------------
// One wave owns a 16-row strip across ALL NT column tiles: A is loaded exactly once per
// k-chunk (v2f per lane), accumulators live in registers (NT x 8 VGPRs), B stays L2-resident.
// K stepped by 4 via V_WMMA_F32_16X16X4_F32.
template <int NT, bool BIAS, bool ACT>
__global__ void gemm_wmma_f32(const float* __restrict__ A, const float* __restrict__ B,
                              const float* __restrict__ bias, float* __restrict__ C, int M) {
  constexpr int K = 64;
  constexpr int Ncol = NT * 16;
  int tm = blockIdx.x * (blockDim.x >> 5) + (threadIdx.x >> 5);  // row-tile per wave
  if (tm * 16 >= M) return;
  int lane = threadIdx.x & 31;
  int lo = lane & 15, hi = lane >> 4;
  int row = tm * 16 + lo;
  const float* arow = A + (size_t)row * K;

  __builtin_prefetch(B + (size_t)lane * 16, 0, 0);   // pull weight matrix toward the WGP
  __builtin_prefetch(arow, 0, 0);

  v8f acc[NT];
#pragma unroll
  for (int t = 0; t < NT; ++t) acc[t] = (v8f){0.f, 0.f, 0.f, 0.f, 0.f, 0.f, 0.f, 0.f};

#pragma unroll 4
  for (int k0 = 0; k0 < K; k0 += 4) {
    int ka = k0 + 2 * hi;                            // this lane's K pair
    v2f a;
    a.x = arow[ka];
    a.y = arow[ka + 1];
    const float* brow = B + (size_t)ka * Ncol;
#pragma unroll
    for (int t = 0; t < NT; ++t) {
      int col = t * 16 + lo;
      v2f b;
      b.x = brow[col];
      b.y = brow[Ncol + col];
      acc[t] = __builtin_amdgcn_wmma_f32_16x16x4_f32(false, a, false, b, (short)0, acc[t],
                                                     false, false);
    }
  }

#pragma unroll
  for (int t = 0; t < NT; ++t) {
    int col = t * 16 + lo;
#pragma unroll
    for (int r = 0; r < 8; ++r) {
      int orow = tm * 16 + r + 8 * hi;
      float v = acc[t][r];
      if (BIAS) v += bias[col];
      if (ACT) v = elu1(v);
      C[(size_t)orow * Ncol + col] = v;
    }
  }
}

// ---------------- fold attention vectors: vs,vd[64,4], ve[12,4]; zero sum4 ----------------
__global__ void fold_att_kernel(const float* __restrict__ Wg, const float* __restrict__ as_,
                                const float* __restrict__ ad_, const float* __restrict__ We,
                                const float* __restrict__ ae_, float* vs, float* vd, float* ve,
                                float* sum4) {
  int t = threadIdx.x;            // 256 threads
  int k = t >> 2, hh = t & 3;
  float s = 0.f, d = 0.f;
  for (int c = 0; c < HIDC; ++c) {
    float w = Wg[k * (NHEAD * HIDC) + hh * HIDC + c];
    s += w * as_[hh * HIDC + c];
    d += w * ad_[hh * HIDC + c];
  }
  vs[t] = s;
  vd[t] = d;
  if (k < EDIM) {
    float e = 0.f;
    for (int c = 0; c < HIDC; ++c) e += We[k * (NHEAD * HIDC) + hh * HIDC + c] * ae_[hh * HIDC + c];
    ve[t] = e;
  }
  if (t < 4) sum4[t] = 0.f;
}

// ---------------- per-node attention logits: al_s/al_d[n,h] = h[n,:] @ vs/vd[:,h] ----------------
__global__ void node_logit_kernel(const float* __restrict__ h, const float* __restrict__ vs,
                                  const float* __restrict__ vd, float* al_s, float* al_d) {
  int tid = blockIdx.x * blockDim.x + threadIdx.x;
  if (tid >= NN * NHEAD) return;
  int n = tid >> 2, hh = tid & 3;
  const float* hr = h + (size_t)n * HIDC;
  float s = 0.f, d = 0.f;
  for (int k = 0; k < HIDC; ++k) {
    float hv = hr[k];
    s += hv * vs[k * 4 + hh];
    d += hv * vd[k * 4 + hh];
  }
  al_s[tid] = s;
  al_d[tid] = d;
}

// ---------------- per-edge logits al_e[e,h] = ea[e,:] @ ve[:,h]; block-reduced sum for loop mean ----------------
__global__ void edge_logit_kernel(const float* __restrict__ ea, const float* __restrict__ ve,
                                  float* al_e, float* sum4) {
  __shared__ float red[256][4];
  int t = threadIdx.x;
  int e = blockIdx.x * 256 + t;
  float v[4] = {0.f, 0.f, 0.f, 0.f};
  if (e < EE) {
    float a[EDIM];
    for (int k = 0; k < EDIM; ++k) a[k] = ea[(size_t)e * EDIM + k];
    for (int hh = 0; hh < 4; ++hh) {
      float acc = 0.f;
      for (int k = 0; k < EDIM; ++k) acc += a[k] * ve[k * 4 + hh];
      v[hh] = acc;
      al_e[(size_t)e * 4 + hh] = acc;
    }
  }
  red[t][0] = v[0]; red[t][1] = v[1]; red[t][2] = v[2]; red[t][3] = v[3];
  __syncthreads();
  for (int s = 128; s > 0; s >>= 1) {
    if (t < s)
      for (int hh = 0; hh < 4; ++hh) red[t][hh] += red[t + s][hh];
    __syncthreads();
  }
  if (t < 4) atomicAdd(&sum4[t], red[0][t]);
}

// ---------------- per-layer init: amax=-inf(enc), denom=0, acc=0 ----------------
__global__ void init_layer_kernel(int* amax, float* denom, float* accb) {
  int tid = blockIdx.x * blockDim.x + threadIdx.x;
  if (tid < NN * NHEAD) { amax[tid] = (int)0x80000000; denom[tid] = 0.f; }
  if (tid < NN * HIDC)  accb[tid] = 0.f;
}

// ---------------- pass A: leaky-relu logits + atomic segment max ----------------
__global__ void attA_kernel(const int* __restrict__ ei, const float* __restrict__ al_s,
                            const float* __restrict__ al_d, const float* __restrict__ al_e,
                            const float* __restrict__ sum4, float* lg, int* amax) {
  int e = blockIdx.x * blockDim.x + threadIdx.x;
  if (e >= ETOT) return;
  int s, d;
  float le[4];
  if (e < EE) {
    s = ei[e]; d = ei[EE + e];
    for (int hh = 0; hh < 4; ++hh) le[hh] = al_e[(size_t)e * 4 + hh];
  } else {
    s = d = e - EE;
    const float inv = 1.0f / (float)EE;
    for (int hh = 0; hh < 4; ++hh) le[hh] = sum4[hh] * inv;   // mean edge_attr logit
  }
  for (int hh = 0; hh < 4; ++hh) {
    float L = al_s[s * 4 + hh] + al_d[d * 4 + hh] + le[hh];
    L = L > 0.f ? L : 0.2f * L;                               // leaky relu
    lg[(size_t)e * 4 + hh] = L;
    atomicMax(&amax[d * 4 + hh], encf(L));
  }
}

// ---------------- pass B: exp(logit - max), atomic denom ----------------
__global__ void attB_kernel(const int* __restrict__ ei, float* lg, const int* __restrict__ amax,
                            float* denom) {
  int idx = blockIdx.x * blockDim.x + threadIdx.x;
  if (idx >= ETOT * NHEAD) return;
  int e = idx >> 2, hh = idx & 3;
  int d = (e < EE) ? ei[EE + e] : (e - EE);
  float ex = expf(lg[idx] - decf(amax[d * 4 + hh]));
  lg[idx] = ex;
  atomicAdd(&denom[d * 4 + hh], ex);
}

// ---------------- pass C: weighted scatter, head-sum folded in ----------------
__global__ void attC_kernel(const int* __restrict__ ei, const float* __restrict__ lg,
                            const float* __restrict__ denom, const float* __restrict__ hp,
                            float* accb) {
  long long tid = (long long)blockIdx.x * blockDim.x + threadIdx.x;
  if (tid >= (long long)ETOT * 32) return;
  int e = (int)(tid >> 5), lane = (int)(tid & 31);
  int s, d;
  if (e < EE) { s = ei[e]; d = ei[EE + e]; } else { s = d = e - EE; }
  float w[4];
  for (int hh = 0; hh < 4; ++hh)
    w[hh] = lg[(size_t)e * 4 + hh] / (denom[d * 4 + hh] + 1e-16f);
  const float* hps = hp + (size_t)s * (NHEAD * HIDC);
  int c0 = lane, c1 = lane + 32;
  float v0 = w[0] * hps[c0] + w[1] * hps[64 + c0] + w[2] * hps[128 + c0] + w[3] * hps[192 + c0];
  float v1 = w[0] * hps[c1] + w[1] * hps[64 + c1] + w[2] * hps[128 + c1] + w[3] * hps[192 + c1];
  atomicAdd(&accb[(size_t)d * HIDC + c0], v0);
  atomicAdd(&accb[(size_t)d * HIDC + c1], v1);
}

// ---------------- layer epilogue: h = elu(acc/H + b) ----------------
__global__ void layer_out_kernel(const float* __restrict__ accb, const float* __restrict__ bg,
                                 float* h) {
  int tid = blockIdx.x * blockDim.x + threadIdx.x;
  if (tid >= NN * HIDC) return;
  int c = tid & (HIDC - 1);
  h[tid] = elu1(accb[tid] * (1.0f / NHEAD) + bg[c]);
}

// ---------------- pooling ----------------
__global__ void pool_init_kernel(float* gsum, float* gcnt) {
  int tid = blockIdx.x * blockDim.x + threadIdx.x;
  if (tid < NGR * HIDC) gsum[tid] = 0.f;
  if (tid < NGR) gcnt[tid] = 0.f;
}

__global__ void pool_scatter_kernel(const float* __restrict__ h, const int* __restrict__ batch,
                                    float* gsum, float* gcnt) {
  int tid = blockIdx.x * blockDim.x + threadIdx.x;
  if (tid >= NN * HIDC) return;
  int n = tid >> 6, c = tid & 63;
  int b = batch[n];
  atomicAdd(&gsum[(size_t)b * HIDC + c], h[tid]);
  if (c == 0) atomicAdd(&gcnt[b], 1.f);
}

// ---------------- per-graph MLP head: out = relu(g@W1+b1)@W2 + b2 ----------------
__global__ void mlp_kernel(const float* __restrict__ gsum, const float* __restrict__ gcnt,
                           const float* __restrict__ W1, const float* __restrict__ b1,
                           const float* __restrict__ W2, const float* __restrict__ b2,
                           float* out) {
  __shared__ float g[64];
  __shared__ float r[64];
  int bg = blockIdx.x, t = threadIdx.x;
  float cnt = gcnt[bg];
  cnt = cnt > 1.f ? cnt : 1.f;
  g[t] = gsum[(size_t)bg * HIDC + t] / cnt;
  __syncthreads();
  float acc = b1[t];
  for (int k = 0; k < HIDC; ++k) acc += g[k] * W1[k * HIDC + t];
  acc = acc > 0.f ? acc : 0.f;
  r[t] = acc * W2[t];
  __syncthreads();
  for (int s = 32; s > 0; s >>= 1) {
    if (t < s) r[t] += r[t + s];
    __syncthreads();
  }
  if (t == 0) out[bg] = r[0] + b2[0];
}

extern "C" void kernel_launch(void* const* d_in, const int* in_sizes, int n_in,
                              void* d_out, int out_size, void* d_ws, size_t ws_size,
                              hipStream_t stream) {
  const float* x        = (const float*)d_in[0];
  const int*   ei       = (const int*)d_in[1];
  const float* ea       = (const float*)d_in[2];
  const int*   batch    = (const int*)d_in[3];
  const float* W_in     = (const float*)d_in[4];
  const float* b_in     = (const float*)d_in[5];
  const float* W_gat    = (const float*)d_in[6];
  const float* att_src  = (const float*)d_in[7];
  const float* att_dst  = (const float*)d_in[8];
  const float* W_edge   = (const float*)d_in[9];
  const float* att_edge = (const float*)d_in[10];
  const float* b_gat    = (const float*)d_in[11];
  const float* W_r1     = (const float*)d_in[12];
  const float* b_r1     = (const float*)d_in[13];
  const float* W_r2     = (const float*)d_in[14];
  const float* b_r2     = (const float*)d_in[15];
  float* out = (float*)d_out;

  // ---- carve workspace (floats) ----
  float* ws = (float*)d_ws;
  size_t o = 0;
  float* h     = ws + o; o += (size_t)NN * HIDC;          // 6.4M
  float* hp    = ws + o; o += (size_t)NN * NHEAD * HIDC;  // 25.6M
  float* accb  = ws + o; o += (size_t)NN * HIDC;          // 6.4M
  float* al_s  = ws + o; o += (size_t)NN * NHEAD;
  float* al_d  = ws + o; o += (size_t)NN * NHEAD;
  float* al_e  = ws + o; o += (size_t)EE * NHEAD;
  float* lg    = ws + o; o += (size_t)ETOT * NHEAD;
  int*   amax  = (int*)(ws + o); o += (size_t)NN * NHEAD;
  float* denom = ws + o; o += (size_t)NN * NHEAD;
  float* vs    = ws + o; o += 256;
  float* vd    = ws + o; o += 256;
  float* ve    = ws + o; o += 64;
  float* sum4  = ws + o; o += 16;
  float* gsum  = ws + o; o += (size_t)NGR * HIDC;
  float* gcnt  = ws + o; o += (size_t)NGR;

  const int TB = 256;
  const int ROW_TILES = NN / 16;               // 6250, exact
  const int WAVES_PER_BLK = TB / 32;           // 8

  // input projection: h = elu(x @ W_in + b_in)
  gemm_wmma_f32<4, true, true><<<CEILDIV(ROW_TILES, WAVES_PER_BLK), TB, 0, stream>>>(
      x, W_in, b_in, h, NN);

  for (int l = 0; l < NLAY; ++l) {
    const float* Wg  = W_gat + (size_t)l * HIDC * NHEAD * HIDC;
    const float* as_ = att_src + (size_t)l * NHEAD * HIDC;
    const float* ad_ = att_dst + (size_t)l * NHEAD * HIDC;
    const float* We  = W_edge + (size_t)l * EDIM * NHEAD * HIDC;
    const float* ae_ = att_edge + (size_t)l * NHEAD * HIDC;
    const float* bg  = b_gat + (size_t)l * HIDC;

    fold_att_kernel<<<1, 256, 0, stream>>>(Wg, as_, ad_, We, ae_, vs, vd, ve, sum4);
    // hp = h @ W_gat[l]  (A read exactly once: one wave covers all 16 column tiles)
    gemm_wmma_f32<16, false, false><<<CEILDIV(ROW_TILES, WAVES_PER_BLK), TB, 0, stream>>>(
        h, Wg, nullptr, hp, NN);
    node_logit_kernel<<<CEILDIV(NN * NHEAD, TB), TB, 0, stream>>>(h, vs, vd, al_s, al_d);
    edge_logit_kernel<<<CEILDIV(EE, TB), TB, 0, stream>>>(ea, ve, al_e, sum4);
    init_layer_kernel<<<CEILDIV(NN * HIDC, TB), TB, 0, stream>>>(amax, denom, accb);
    attA_kernel<<<CEILDIV(ETOT, TB), TB, 0, stream>>>(ei, al_s, al_d, al_e, sum4, lg, amax);
    attB_kernel<<<CEILDIV(ETOT * NHEAD, TB), TB, 0, stream>>>(ei, lg, amax, denom);
    attC_kernel<<<(int)CEILDIV((long long)ETOT * 32, TB), TB, 0, stream>>>(ei, lg, denom, hp, accb);
    layer_out_kernel<<<CEILDIV(NN * HIDC, TB), TB, 0, stream>>>(accb, bg, h);
  }

  pool_init_kernel<<<CEILDIV(NGR * HIDC, TB), TB, 0, stream>>>(gsum, gcnt);
  pool_scatter_kernel<<<CEILDIV(NN * HIDC, TB), TB, 0, stream>>>(h, batch, gsum, gcnt);
  mlp_kernel<<<NGR, 64, 0, stream>>>(gsum, gcnt, W_r1, b_r1, W_r2, b_r2, out);
}